// TGCNCell_6966436954285
// MI455X (gfx1250) — compile-verified
//
#include <hip/hip_runtime.h>
#include <cmath>

#define N_NODES 50000
#define N_EDGES 800000

typedef __attribute__((ext_vector_type(2))) float v2f;
typedef __attribute__((ext_vector_type(8))) float v8f;

__device__ __forceinline__ void atomic_add_f32(float* p, float v) {
  // lowers to global_atomic_add_f32 (relaxed, device scope) on gfx1250
  __hip_atomic_fetch_add(p, v, __ATOMIC_RELAXED, __HIP_MEMORY_SCOPE_AGENT);
}

// ---------------- degree / dinv ----------------
__global__ __launch_bounds__(256) void k_deg_init(float* __restrict__ deg, int n) {
  int i = blockIdx.x * blockDim.x + threadIdx.x;
  if (i < n) deg[i] = 1.0f;  // self-loop weight
}

__global__ __launch_bounds__(256) void k_deg_accum(const int* __restrict__ ei,
                                                   const float* __restrict__ ew,
                                                   float* __restrict__ deg, int e_cnt) {
  int e = blockIdx.x * blockDim.x + threadIdx.x;
  if (e < e_cnt) {
    int dst = ei[e_cnt + e];  // edge_index row 1
    atomic_add_f32(&deg[dst], ew[e]);
  }
}

__global__ __launch_bounds__(256) void k_deg_finish(float* __restrict__ deg, int n) {
  int i = blockIdx.x * blockDim.x + threadIdx.x;
  if (i < n) deg[i] = rsqrtf(deg[i]);  // deg >= 1 always
}

// ---------------- WMMA GEMM: OUT[n,16*nct] = [X|Hp] @ (Wa|Wb) ----------------
// Virtual A concat: cols 0..63 from X, 64..127 from Hp. Col-tiles 0..3 -> Wa, 4..7 -> Wb.
// One wave computes one 16x16 f32 tile via 32x V_WMMA_F32_16X16X4_F32.
__global__ __launch_bounds__(256) void k_gemm_wmma(const float* __restrict__ X,
                                                   const float* __restrict__ Hp,
                                                   const float* __restrict__ Wa,
                                                   const float* __restrict__ Wb,
                                                   float* __restrict__ OUT,
                                                   int n_rowtiles, int n_coltiles) {
  int wave = (int)((blockIdx.x * blockDim.x + threadIdx.x) >> 5);
  int lane = threadIdx.x & 31;
  if (wave >= n_rowtiles * n_coltiles) return;  // wave-uniform exit (EXEC all-1s for WMMA)

  int mt = wave / n_coltiles;
  int ct = wave % n_coltiles;
  const float* W = (ct < 4) ? Wa : Wb;
  int wcol = ((ct < 4) ? ct : ct - 4) * 16;

  int half = lane >> 4;  // 0 or 1 (K sub-pair select)
  int m    = lane & 15;
  int row  = mt * 16 + m;   // A row handled by this lane
  int col  = wcol + m;      // B column handled by this lane
  const float* arow_x = X  + (size_t)row * 64;
  const float* arow_h = Hp + (size_t)row * 64;

  v8f acc = {};
#pragma unroll
  for (int kk = 0; kk < 128; kk += 4) {
    int k0 = kk + half * 2;  // pair (k0, k0+1) never straddles the 64 boundary
    v2f a = (kk < 64) ? *(const v2f*)(arow_x + k0)
                      : *(const v2f*)(arow_h + (k0 - 64));
    v2f b;
    b.x = W[(size_t)k0 * 64 + col];
    b.y = W[(size_t)(k0 + 1) * 64 + col];
    acc = __builtin_amdgcn_wmma_f32_16x16x4_f32(false, a, false, b, (short)0, acc,
                                                false, false);
  }

  int ow   = n_coltiles * 16;
  int ocol = ct * 16 + m;
#pragma unroll
  for (int v = 0; v < 8; ++v) {
    int orow = mt * 16 + v + 8 * half;  // C/D layout: lanes16-31 hold rows M+8
    OUT[(size_t)orow * ow + ocol] = acc[v];
  }
}

// ---------------- aggregation: self-loop init then edge scatter ----------------
__global__ __launch_bounds__(256) void k_agg_init(const float* __restrict__ H,
                                                  const float* __restrict__ dinv,
                                                  const float* __restrict__ b1,
                                                  const float* __restrict__ b2,
                                                  float* __restrict__ A, int n, int ow) {
  int t = blockIdx.x * blockDim.x + threadIdx.x;
  if (t >= n * ow) return;
  int node = t / ow;
  int c    = t % ow;
  float bias = (c < 64) ? b1[c] : b2[c - 64];
  float di   = dinv[node];
  A[t] = H[t] * di * di + bias;
}

__global__ __launch_bounds__(256) void k_edge_scatter(const int* __restrict__ ei,
                                                      const float* __restrict__ ew,
                                                      const float* __restrict__ dinv,
                                                      const float* __restrict__ H,
                                                      float* __restrict__ A,
                                                      int e_cnt, int ow) {
  int t   = blockIdx.x * blockDim.x + threadIdx.x;
  int tpe = ow >> 2;  // threads per edge, 4 channels each
  int e   = t / tpe;
  if (e >= e_cnt) return;
  int c   = (t % tpe) * 4;
  int src = ei[e];
  int dst = ei[e_cnt + e];
  float norm = dinv[src] * ew[e] * dinv[dst];
  const float4 hv = *(const float4*)(H + (size_t)src * ow + c);
  float* ap = A + (size_t)dst * ow + c;
  atomic_add_f32(ap + 0, hv.x * norm);
  atomic_add_f32(ap + 1, hv.y * norm);
  atomic_add_f32(ap + 2, hv.z * norm);
  atomic_add_f32(ap + 3, hv.w * norm);
}

// ---------------- gates ----------------
__global__ __launch_bounds__(256) void k_gates(const float* __restrict__ A /*Nx128*/,
                                               const float* __restrict__ h,
                                               float* __restrict__ u,
                                               float* __restrict__ rh, int n64) {
  int t = blockIdx.x * blockDim.x + threadIdx.x;
  if (t >= n64) return;
  int node = t >> 6;
  int c    = t & 63;
  float rv = A[(size_t)node * 128 + c];
  float uv = A[(size_t)node * 128 + 64 + c];
  rv = 1.0f / (1.0f + __expf(-rv));
  uv = 1.0f / (1.0f + __expf(-uv));
  u[t]  = uv;
  rh[t] = rv * h[t];
}

__global__ __launch_bounds__(256) void k_final(const float* __restrict__ A3 /*Nx64*/,
                                               const float* __restrict__ u,
                                               const float* __restrict__ h,
                                               float* __restrict__ out, int n64) {
  int t = blockIdx.x * blockDim.x + threadIdx.x;
  if (t >= n64) return;
  float cc = tanhf(A3[t]);
  float uv = u[t];
  out[t] = uv * h[t] + (1.0f - uv) * cc;
}

extern "C" void kernel_launch(void* const* d_in, const int* in_sizes, int n_in,
                              void* d_out, int out_size, void* d_ws, size_t ws_size,
                              hipStream_t stream) {
  (void)in_sizes; (void)n_in; (void)out_size; (void)ws_size;
  const float* x  = (const float*)d_in[0];
  const int*   ei = (const int*)d_in[1];   // [2,E] row-major: src then dst
  const float* ew = (const float*)d_in[2];
  const float* h  = (const float*)d_in[3];
  const float* W1 = (const float*)d_in[4];
  const float* b1 = (const float*)d_in[5];
  const float* W2 = (const float*)d_in[6];
  const float* b2 = (const float*)d_in[7];
  const float* W3 = (const float*)d_in[8];
  const float* b3 = (const float*)d_in[9];
  float* out = (float*)d_out;

  const int N = N_NODES, E = N_EDGES;
  float* ws   = (float*)d_ws;
  float* dinv = ws;                         // [N]      degree -> dinv in place
  float* Hbuf = ws + (size_t)N;             // [N,128]  H1|H2, later H3 in [N,64]
  float* Abuf = Hbuf + (size_t)N * 128;     // [N,128]  AGG1|AGG2, later AGG3 [N,64]
  float* ubuf = Abuf + (size_t)N * 128;     // [N,64]
  float* rhbf = ubuf + (size_t)N * 64;      // [N,64]

  const int B = 256;
  const int rowtiles = N / 16;  // 3125, exact

  // degree / dinv
  k_deg_init  <<<(N + B - 1) / B, B, 0, stream>>>(dinv, N);
  k_deg_accum <<<(E + B - 1) / B, B, 0, stream>>>(ei, ew, dinv, E);
  k_deg_finish<<<(N + B - 1) / B, B, 0, stream>>>(dinv, N);

  // H1|H2 = [x,h] @ [W1|W2]   (8 col-tiles -> OW=128)
  {
    int waves = rowtiles * 8;
    k_gemm_wmma<<<(waves * 32 + B - 1) / B, B, 0, stream>>>(x, h, W1, W2, Hbuf,
                                                            rowtiles, 8);
  }
  // AGG = self-loop + bias, then edge scatter
  k_agg_init<<<((size_t)N * 128 + B - 1) / B, B, 0, stream>>>(Hbuf, dinv, b1, b2,
                                                              Abuf, N, 128);
  k_edge_scatter<<<((size_t)E * 32 + B - 1) / B, B, 0, stream>>>(ei, ew, dinv, Hbuf,
                                                                 Abuf, E, 128);
  // r,u gates; rh = r*h
  k_gates<<<((size_t)N * 64 + B - 1) / B, B, 0, stream>>>(Abuf, h, ubuf, rhbf,
                                                          N * 64);

  // H3 = [x, rh] @ W3   (4 col-tiles -> OW=64), reuse Hbuf
  {
    int waves = rowtiles * 4;
    k_gemm_wmma<<<(waves * 32 + B - 1) / B, B, 0, stream>>>(x, rhbf, W3, W3, Hbuf,
                                                            rowtiles, 4);
  }
  // AGG3, reuse Abuf
  k_agg_init<<<((size_t)N * 64 + B - 1) / B, B, 0, stream>>>(Hbuf, dinv, b3, b3,
                                                             Abuf, N, 64);
  k_edge_scatter<<<((size_t)E * 16 + B - 1) / B, B, 0, stream>>>(ei, ew, dinv, Hbuf,
                                                                 Abuf, E, 64);
  // h_new = u*h + (1-u)*tanh(AGG3)
  k_final<<<((size_t)N * 64 + B - 1) / B, B, 0, stream>>>(Abuf, ubuf, h, out, N * 64);
}